// MyLoss_31078383354025
// MI455X (gfx1250) — compile-verified
//
#include <hip/hip_runtime.h>
#include <hip/hip_bf16.h>
#include <math.h>

typedef __attribute__((ext_vector_type(2))) float v2f;
typedef __attribute__((ext_vector_type(8))) float v8f;

// Normalized 1D Gaussian, WIN=11, sigma=1.5 (matches reference f32 values to ~1e-7)
__device__ __constant__ float c_g[11] = {
    0.00102838f, 0.00759877f, 0.03600077f, 0.10936151f, 0.21300553f,
    0.26601008f, 0.21300553f, 0.10936151f, 0.03600077f, 0.00759877f,
    0.00102838f};

#define IMGW 2048
#define NPIX (2048 * 2048)

// ---------------- deltaE2000 ----------------

__device__ __forceinline__ float srgb_lin(float c) {
    return (c <= 0.04045f) ? c * (1.0f / 12.92f)
                           : __powf((c + 0.055f) * (1.0f / 1.055f), 2.4f);
}

__device__ __forceinline__ void rgb2lab(float r, float g, float b,
                                        float& L, float& A, float& B) {
    float rl = srgb_lin(r), gl = srgb_lin(g), bl = srgb_lin(b);
    float x = (0.4124564f * rl + 0.3575761f * gl + 0.1804375f * bl) * (1.0f / 0.95047f);
    float y = (0.2126729f * rl + 0.7151522f * gl + 0.0721750f * bl);
    float z = (0.0193339f * rl + 0.1191920f * gl + 0.9503041f * bl) * (1.0f / 1.08883f);
    const float eps = 0.008856451679036f;   // (6/29)^3
    const float kk  = 7.787037037037f;      // 1/(3*(6/29)^2)
    const float c16 = 4.0f / 29.0f;
    float fx = (x > eps) ? __powf(x, 0.33333333333333f) : fmaf(kk, x, c16);
    float fy = (y > eps) ? __powf(y, 0.33333333333333f) : fmaf(kk, y, c16);
    float fz = (z > eps) ? __powf(z, 0.33333333333333f) : fmaf(kk, z, c16);
    L = 116.0f * fy - 16.0f;
    A = 500.0f * (fx - fy);
    B = 200.0f * (fy - fz);
}

__device__ __forceinline__ float pow7f(float x) {
    float x2 = x * x;
    float x3 = x2 * x;
    return x3 * x3 * x;
}

__device__ float dE2000(float L1, float a1, float b1,
                        float L2, float a2, float b2) {
    const float d2r = 0.017453292519943295f;
    const float r2d = 57.29577951308232f;
    const float P25_7 = 6103515625.0f;  // 25^7

    float C1 = sqrtf(a1 * a1 + b1 * b1);
    float C2 = sqrtf(a2 * a2 + b2 * b2);
    float Cbar = 0.5f * (C1 + C2);
    float Cbar7 = pow7f(Cbar);
    float G = 0.5f * (1.0f - sqrtf(Cbar7 / (Cbar7 + P25_7)));
    float a1p = (1.0f + G) * a1;
    float a2p = (1.0f + G) * a2;
    float C1p = sqrtf(a1p * a1p + b1 * b1);
    float C2p = sqrtf(a2p * a2p + b2 * b2);

    float h1p = atan2f(b1, a1p) * r2d;
    h1p = fmodf(h1p, 360.0f); if (h1p < 0.0f) h1p += 360.0f;
    float h2p = atan2f(b2, a2p) * r2d;
    h2p = fmodf(h2p, 360.0f); if (h2p < 0.0f) h2p += 360.0f;

    float dLp = L2 - L1;
    float dCp = C2p - C1p;
    float dh = h2p - h1p;
    if (dh > 180.0f) dh -= 360.0f;
    else if (dh < -180.0f) dh += 360.0f;
    float CC = C1p * C2p;
    if (CC == 0.0f) dh = 0.0f;
    float dHp = 2.0f * sqrtf(CC) * __sinf(0.5f * dh * d2r);

    float Lbar = 0.5f * (L1 + L2);
    float Cbarp = 0.5f * (C1p + C2p);
    float hsum = h1p + h2p;
    float hbar;
    if (fabsf(h1p - h2p) <= 180.0f) hbar = 0.5f * hsum;
    else if (hsum < 360.0f) hbar = 0.5f * (hsum + 360.0f);
    else hbar = 0.5f * (hsum - 360.0f);
    if (CC == 0.0f) hbar = hsum;

    float T = 1.0f - 0.17f * __cosf((hbar - 30.0f) * d2r)
                   + 0.24f * __cosf(2.0f * hbar * d2r)
                   + 0.32f * __cosf((3.0f * hbar + 6.0f) * d2r)
                   - 0.20f * __cosf((4.0f * hbar - 63.0f) * d2r);
    float hx = (hbar - 275.0f) * (1.0f / 25.0f);
    float dtheta = 30.0f * __expf(-hx * hx);
    float Cbarp7 = pow7f(Cbarp);
    float RC = 2.0f * sqrtf(Cbarp7 / (Cbarp7 + P25_7));
    float Lm50 = Lbar - 50.0f;
    float Lm50sq = Lm50 * Lm50;
    float SL = 1.0f + 0.015f * Lm50sq * rsqrtf(20.0f + Lm50sq);
    float SC = 1.0f + 0.045f * Cbarp;
    float SH = 1.0f + 0.015f * Cbarp * T;
    float RT = -__sinf(2.0f * dtheta * d2r) * RC;
    float tL = dLp / SL, tC = dCp / SC, tH = dHp / SH;
    return sqrtf(tL * tL + tC * tC + tH * tH + RT * tC * tH);
}

__global__ __launch_bounds__(256) void de_kernel(const float* __restrict__ a,
                                                 const float* __restrict__ b,
                                                 double* __restrict__ ws) {
    float lsum = 0.0f;
    int stride = (int)(gridDim.x * blockDim.x);
    for (int i = (int)(blockIdx.x * blockDim.x + threadIdx.x); i < NPIX; i += stride) {
        float L1, A1, B1, L2, A2, B2;
        rgb2lab(a[i], a[i + NPIX], a[i + 2 * NPIX], L1, A1, B1);
        rgb2lab(b[i], b[i + NPIX], b[i + 2 * NPIX], L2, A2, B2);
        lsum += dE2000(L1, A1, B1, L2, A2, B2);
    }
    // wave32 reduce, then cross-wave via LDS
    for (int off = 16; off > 0; off >>= 1) lsum += __shfl_down(lsum, off, 32);
    __shared__ float red[8];
    int lane = threadIdx.x & 31, wid = threadIdx.x >> 5;
    if (lane == 0) red[wid] = lsum;
    __syncthreads();
    if (threadIdx.x == 0) {
        float t = 0.0f;
        for (int i = 0; i < 8; ++i) t += red[i];
        atomicAdd(ws, (double)t);
    }
}

// ---------------- SSIM (fused tile kernel, vertical pass via f32 WMMA) ----------------

#define TS 32       // output tile (32x32)
#define IR 42       // input region = TS + 10 halo
#define IPITCH 43
#define HR 44       // horizontal-pass rows (42 + 2 zero pad so K chunks of 4 are safe)
#define HPITCH 32

struct SsimShared {
    float in1[IR][IPITCH];
    float in2[IR][IPITCH];
    float h[5][HR][HPITCH];   // blurred-horizontally: x, y, x^2, y^2, xy
    float out[5][TS][TS];
    float sg[11];
    float red[8];
};

__global__ __launch_bounds__(256) void ssim_kernel(const float* __restrict__ img1,
                                                   const float* __restrict__ img2,
                                                   double* __restrict__ ws) {
    __shared__ SsimShared sm;
    const int tid = (int)threadIdx.x;
    const int tilesPerRow = IMGW / TS;                 // 64
    const int tilesPerCh = tilesPerRow * tilesPerRow;  // 4096
    const int bid = (int)blockIdx.x;                   // 0 .. 3*4096-1
    const int ch = bid / tilesPerCh;
    const int t = bid % tilesPerCh;
    const int tr = t / tilesPerRow, tc = t % tilesPerRow;
    const int r0g = tr * TS - 5, c0g = tc * TS - 5;    // input-region origin (zero pad)
    const float* p1 = img1 + (size_t)ch * NPIX;
    const float* p2 = img2 + (size_t)ch * NPIX;

    if (tid < 11) sm.sg[tid] = c_g[tid];

    // Stage input tiles (zero-padded boundary)
    for (int idx = tid; idx < IR * IR; idx += 256) {
        int r = idx / IR, c = idx % IR;
        int gr = r0g + r, gc = c0g + c;
        bool ok = ((unsigned)gr < (unsigned)IMGW) && ((unsigned)gc < (unsigned)IMGW);
        size_t off = (size_t)gr * IMGW + (size_t)gc;
        sm.in1[r][c] = ok ? p1[off] : 0.0f;
        sm.in2[r][c] = ok ? p2[off] : 0.0f;
    }
    __syncthreads();

    // Horizontal Gaussian over the 5 products
    for (int idx = tid; idx < IR * TS; idx += 256) {
        int r = idx / TS, c = idx % TS;
        float s0 = 0.f, s1 = 0.f, s2 = 0.f, s3 = 0.f, s4 = 0.f;
#pragma unroll
        for (int k = 0; k < 11; ++k) {
            float g = c_g[k];
            float xv = sm.in1[r][c + k];
            float yv = sm.in2[r][c + k];
            s0 = fmaf(g, xv, s0);
            s1 = fmaf(g, yv, s1);
            s2 = fmaf(g, xv * xv, s2);
            s3 = fmaf(g, yv * yv, s3);
            s4 = fmaf(g, xv * yv, s4);
        }
        sm.h[0][r][c] = s0; sm.h[1][r][c] = s1; sm.h[2][r][c] = s2;
        sm.h[3][r][c] = s3; sm.h[4][r][c] = s4;
    }
    // Zero rows 42,43 (read by the padded K-chunks, multiplied by zero weights)
    for (int idx = tid; idx < 5 * 2 * TS; idx += 256) {
        int p = idx / (2 * TS);
        int rem = idx % (2 * TS);
        sm.h[p][IR + rem / TS][rem % TS] = 0.0f;
    }
    __syncthreads();

    // Vertical pass as banded matmul: Out(32x32) = W(32x44) * H(44x32),
    // W[r][j] = g[j-r] for 0<=j-r<=10 else 0.  V_WMMA_F32_16X16X4_F32, K chunks of 4.
    const int lane = tid & 31, wid = tid >> 5;
    const int m16 = lane & 15;       // row (A) / col (B,D) within 16
    const int hh = lane >> 4;        // half-wave selector

    // A fragments: a[i].x = W[r0+m16][r0+4i+2*hh], independent of r0 (banded)
    v2f afrag[7];
#pragma unroll
    for (int i = 0; i < 7; ++i) {
        int d0 = 4 * i + 2 * hh - m16;
        int d1 = d0 + 1;
        afrag[i].x = (d0 >= 0 && d0 <= 10) ? sm.sg[d0] : 0.0f;
        afrag[i].y = (d1 >= 0 && d1 <= 10) ? sm.sg[d1] : 0.0f;
    }

    // 20 jobs = 5 products x 4 (16x16) output tiles, round-robin over 8 waves
    for (int job = wid; job < 20; job += 8) {
        int p = job % 5;
        int tt = job / 5;              // 0..3
        int r0 = (tt >> 1) * 16;       // output row block
        int n0 = (tt & 1) * 16;        // output col block
        v8f acc = {0.f, 0.f, 0.f, 0.f, 0.f, 0.f, 0.f, 0.f};
#pragma unroll
        for (int i = 0; i < 7; ++i) {
            int k0 = r0 + 4 * i + 2 * hh;
            v2f b;
            b.x = sm.h[p][k0][n0 + m16];
            b.y = sm.h[p][k0 + 1][n0 + m16];
            acc = __builtin_amdgcn_wmma_f32_16x16x4_f32(
                false, afrag[i], false, b, (short)0, acc, false, false);
        }
#pragma unroll
        for (int v = 0; v < 8; ++v)
            sm.out[p][r0 + v + 8 * hh][n0 + m16] = acc[v];
    }
    __syncthreads();

    // Per-pixel SSIM map + block reduction
    float lsum = 0.0f;
    for (int idx = tid; idx < TS * TS; idx += 256) {
        int r = idx / TS, c = idx % TS;
        float mu1 = sm.out[0][r][c], mu2 = sm.out[1][r][c];
        float ex2 = sm.out[2][r][c], ey2 = sm.out[3][r][c], exy = sm.out[4][r][c];
        float mu1sq = mu1 * mu1, mu2sq = mu2 * mu2, mu12 = mu1 * mu2;
        float sg1 = ex2 - mu1sq, sg2 = ey2 - mu2sq, sg12 = exy - mu12;
        const float C1 = 0.0001f;   // 0.01^2
        const float C2 = 0.0009f;   // 0.03^2
        float num = (2.0f * mu12 + C1) * (2.0f * sg12 + C2);
        float den = (mu1sq + mu2sq + C1) * (sg1 + sg2 + C2);
        lsum += num / den;
    }
    for (int off = 16; off > 0; off >>= 1) lsum += __shfl_down(lsum, off, 32);
    if (lane == 0) sm.red[wid] = lsum;
    __syncthreads();
    if (tid == 0) {
        float tot = 0.0f;
        for (int i = 0; i < 8; ++i) tot += sm.red[i];
        atomicAdd(ws + 1, (double)tot);
    }
}

// ---------------- scalar finalize + workspace init ----------------

__global__ void zero_kernel(double* ws) {
    if (threadIdx.x == 0) { ws[0] = 0.0; ws[1] = 0.0; }
}

__global__ void fin_kernel(const double* __restrict__ ws, float* __restrict__ out) {
    if (threadIdx.x == 0 && blockIdx.x == 0) {
        double de = ws[0] / ((double)NPIX);
        float s = (float)(ws[1] / (3.0 * (double)NPIX));
        float st;
        if (s < 0.0f) {
            float u = 1.0f - fabsf(s);
            st = u * u * 0.01f;
        } else if (s == 0.0f) {
            st = 0.01f;
        } else {
            st = s;
        }
        out[0] = (float)(de / (double)st);
    }
}

extern "C" void kernel_launch(void* const* d_in, const int* in_sizes, int n_in,
                              void* d_out, int out_size, void* d_ws, size_t ws_size,
                              hipStream_t stream) {
    const float* input = (const float*)d_in[0];
    const float* target = (const float*)d_in[1];
    double* ws = (double*)d_ws;
    float* out = (float*)d_out;

    zero_kernel<<<1, 64, 0, stream>>>(ws);
    de_kernel<<<4096, 256, 0, stream>>>(input, target, ws);
    ssim_kernel<<<3 * (IMGW / TS) * (IMGW / TS), 256, 0, stream>>>(input, target, ws);
    fin_kernel<<<1, 64, 0, stream>>>(ws, out);
}